// CounterfactualReasoning_37563783971144
// MI455X (gfx1250) — compile-verified
//
#include <hip/hip_runtime.h>

typedef float v2f __attribute__((ext_vector_type(2)));
typedef float v8f __attribute__((ext_vector_type(8)));

#define BROWS 4096
#define DCOLS 512
#define NSTEPS 50

// Each block: 16 rows, 256 threads = 8 waves; wave w owns columns [w*64, w*64+64).
// Per-lane register state in WMMA A-matrix layout:
//   lane L -> row M = L&15 ; K within a 4-wide chunk = 2*(L>>4) + j, j in {0,1}
// so chunk c, reg j holds element (row, wave*64 + c*4 + 2*(L>>4) + j).
__global__ __launch_bounds__(256) void adam_wmma_kernel(
    const float* __restrict__ S, const float* __restrict__ T,
    const float* __restrict__ Wt, const float* __restrict__ Bs,
    float* __restrict__ Out)
{
    __shared__ float zpart[8][16];

    const int tid  = threadIdx.x;
    const int wave = tid >> 5;
    const int lane = tid & 31;
    const int hf   = lane >> 4;   // half-wave (K offset 0 or 2)
    const int mrow = lane & 15;   // A-matrix row within tile

    const int row = blockIdx.x * 16 + mrow;
    const int cb  = wave * 64;

    float ax[32], am[32], av[32], as_[32], aw[32];

    #pragma unroll
    for (int c = 0; c < 16; ++c) {
        const int k0 = cb + c * 4 + hf * 2;            // even -> 8B aligned
        const float2 sv = *reinterpret_cast<const float2*>(S + (size_t)row * DCOLS + k0);
        const float2 wv = *reinterpret_cast<const float2*>(Wt + k0);
        ax[2*c] = sv.x;  ax[2*c+1] = sv.y;             // x init = s
        as_[2*c] = sv.x; as_[2*c+1] = sv.y;
        aw[2*c] = wv.x;  aw[2*c+1] = wv.y;
        am[2*c] = 0.f;   am[2*c+1] = 0.f;
        av[2*c] = 0.f;   av[2*c+1] = 0.f;
    }
    const float tval = T[row];
    const float bval = Bs[0];

    const float C2B    = 2.0f / (float)BROWS;                    // d(mse)/dz scale
    const float INV_BD = 1.0f / ((float)BROWS * (float)DCOLS);   // d(l1)/dx scale
    const float LR = 0.01f, B1 = 0.9f, B2 = 0.999f, EPS = 1e-8f;
    float pb1 = 1.f, pb2 = 1.f;                                  // B1^t, B2^t

    for (int step = 0; step < NSTEPS; ++step) {
        // z partials for this wave's 64 columns on the matrix core.
        // 4 independent accumulator chains (4 deep) to hide WMMA pipe latency.
        v8f acc0 = {0.f,0.f,0.f,0.f,0.f,0.f,0.f,0.f};
        v8f acc1 = {0.f,0.f,0.f,0.f,0.f,0.f,0.f,0.f};
        v8f acc2 = {0.f,0.f,0.f,0.f,0.f,0.f,0.f,0.f};
        v8f acc3 = {0.f,0.f,0.f,0.f,0.f,0.f,0.f,0.f};
        #pragma unroll
        for (int c = 0; c < 16; c += 4) {
            v2f a0, b0, a1, b1, a2, b2, a3, b3;
            a0.x = ax[2*c+0]; a0.y = ax[2*c+1]; b0.x = aw[2*c+0]; b0.y = aw[2*c+1];
            a1.x = ax[2*c+2]; a1.y = ax[2*c+3]; b1.x = aw[2*c+2]; b1.y = aw[2*c+3];
            a2.x = ax[2*c+4]; a2.y = ax[2*c+5]; b2.x = aw[2*c+4]; b2.y = aw[2*c+5];
            a3.x = ax[2*c+6]; a3.y = ax[2*c+7]; b3.x = aw[2*c+6]; b3.y = aw[2*c+7];
            acc0 = __builtin_amdgcn_wmma_f32_16x16x4_f32(false, a0, false, b0, (short)0, acc0, false, false);
            acc1 = __builtin_amdgcn_wmma_f32_16x16x4_f32(false, a1, false, b1, (short)0, acc1, false, false);
            acc2 = __builtin_amdgcn_wmma_f32_16x16x4_f32(false, a2, false, b2, (short)0, acc2, false, false);
            acc3 = __builtin_amdgcn_wmma_f32_16x16x4_f32(false, a3, false, b3, (short)0, acc3, false, false);
        }
        const v8f acc = (acc0 + acc1) + (acc2 + acc3);

        // C layout: reg r, lanes 0-15 -> (M=r, N=lane); lanes 16-31 -> (M=r+8, N=lane-16).
        // z replicated over N, so lanes with N==0 (lane 0 and 16) publish 8 rows each.
        if (mrow == 0) {
            #pragma unroll
            for (int r = 0; r < 8; ++r) zpart[wave][hf * 8 + r] = acc[r];
        }
        __syncthreads();

        float z = bval;
        #pragma unroll
        for (int w = 0; w < 8; ++w) z += zpart[w][mrow];
        const float p    = __builtin_amdgcn_rcpf(1.0f + __expf(-z));
        const float coef = C2B * (p - tval) * p * (1.0f - p);

        pb1 *= B1; pb2 *= B2;
        const float lrc1 = LR * __builtin_amdgcn_rcpf(1.0f - pb1);  // LR / (1-B1^t)
        const float c2   = __builtin_amdgcn_rcpf(1.0f - pb2);       // 1 / (1-B2^t)

        #pragma unroll
        for (int i = 0; i < 32; ++i) {
            const float d  = ax[i] - as_[i];
            const float sg = (d > 0.f) ? 1.f : ((d < 0.f) ? -1.f : 0.f);
            const float g  = coef * aw[i] + sg * INV_BD;
            am[i] = B1 * am[i] + (1.f - B1) * g;
            av[i] = B2 * av[i] + (1.f - B2) * g * g;
            const float denom = __builtin_amdgcn_sqrtf(av[i] * c2) + EPS;
            ax[i] -= lrc1 * am[i] * __builtin_amdgcn_rcpf(denom);
        }
        __syncthreads();   // zpart safe to overwrite next step
    }

    #pragma unroll
    for (int c = 0; c < 16; ++c) {
        const int k0 = cb + c * 4 + hf * 2;
        float2 ov; ov.x = ax[2*c]; ov.y = ax[2*c+1];
        *reinterpret_cast<float2*>(Out + (size_t)row * DCOLS + k0) = ov;
    }
}

extern "C" void kernel_launch(void* const* d_in, const int* in_sizes, int n_in,
                              void* d_out, int out_size, void* d_ws, size_t ws_size,
                              hipStream_t stream) {
    const float* S  = (const float*)d_in[0];   // s            [4096, 512]
    const float* T  = (const float*)d_in[1];   // target_conf  [4096, 1]
    const float* Wt = (const float*)d_in[2];   // W            [1, 512]
    const float* Bs = (const float*)d_in[3];   // b            [1]
    float* Out = (float*)d_out;                // adjusted_s   [4096, 512]

    dim3 grid(BROWS / 16);
    dim3 block(256);
    adam_wmma_kernel<<<grid, block, 0, stream>>>(S, T, Wt, Bs, Out);
}